// WindowedSelfAttention_28174985462417
// MI455X (gfx1250) — compile-verified
//
#include <hip/hip_runtime.h>
#include <hip/hip_bf16.h>

typedef __attribute__((ext_vector_type(2))) float v2f;
typedef __attribute__((ext_vector_type(8))) float v8f;

#define BDIM 2
#define TDIM 256
#define NDIM 8
#define EDIM 512
#define HEADS 8
#define DHEAD 64
#define TN 2048              // T*N rows per batch
#define BAND 272             // 8*(33+1): key band for a 2-timestep query tile
#define QKV_ELEMS (2u*8u*2048u*64u)   // per-projection workspace elems

// ---------------------------------------------------------------------------
// Projection GEMM: out[p][b][h][r][dc] = (X_p[row] . W_p[col]) + bias, scaled.
// One wave computes a 16x64 output tile with V_WMMA_F32_16X16X4_F32.
// ---------------------------------------------------------------------------
__global__ __launch_bounds__(256) void proj_kernel(
    const float* __restrict__ Xq, const float* __restrict__ Xk, const float* __restrict__ Xv,
    const float* __restrict__ Wq, const float* __restrict__ bq,
    const float* __restrict__ Wk, const float* __restrict__ bk,
    const float* __restrict__ Wv, const float* __restrict__ bv,
    float* __restrict__ ws)
{
    const int lane = threadIdx.x & 31;
    const int lm = lane & 15;     // row-in-tile (A) / col-in-tile (B,D)
    const int lh = lane >> 4;     // K-slice half select

    const int wid = blockIdx.x * (blockDim.x >> 5) + (threadIdx.x >> 5);  // 0..6143
    const int p   = wid >> 11;            // 0=q,1=k,2=v
    const int rem = wid & 2047;
    const int j64 = rem & 7;              // 64-wide column group == head
    const int mt  = rem >> 3;             // 0..255
    const int r0  = mt << 4;              // global row (0..4095, step 16)

    const float* X    = (p == 0) ? Xq : (p == 1) ? Xk : Xv;
    const float* Wm   = (p == 0) ? Wq : (p == 1) ? Wk : Wv;
    const float* bias = (p == 0) ? bq : (p == 1) ? bk : bv;
    float* out = ws + (size_t)p * QKV_ELEMS;
    const float scale = (p == 0) ? 0.125f : 1.0f;   // q /= sqrt(64)

    const int j0 = j64 << 6;

    // A: lane reads X[r0+lm][e0 + 2*lh + {0,1}]
    const float* xrow  = X  + (size_t)(r0 + lm) * EDIM + 2 * lh;
    // B[e][j] = W[j][e]: lane reads W[j0+ntile*16+lm][e0 + 2*lh + {0,1}]
    const float* wrow0 = Wm + (size_t)(j0 +  0 + lm) * EDIM + 2 * lh;
    const float* wrow1 = Wm + (size_t)(j0 + 16 + lm) * EDIM + 2 * lh;
    const float* wrow2 = Wm + (size_t)(j0 + 32 + lm) * EDIM + 2 * lh;
    const float* wrow3 = Wm + (size_t)(j0 + 48 + lm) * EDIM + 2 * lh;

    v8f acc0 = {}, acc1 = {}, acc2 = {}, acc3 = {};
    for (int e0 = 0; e0 < EDIM; e0 += 4) {
        v2f a  = *(const v2f*)(xrow  + e0);
        v2f b0 = *(const v2f*)(wrow0 + e0);
        v2f b1 = *(const v2f*)(wrow1 + e0);
        v2f b2 = *(const v2f*)(wrow2 + e0);
        v2f b3 = *(const v2f*)(wrow3 + e0);
        acc0 = __builtin_amdgcn_wmma_f32_16x16x4_f32(false, a, false, b0, (short)0, acc0, false, false);
        acc1 = __builtin_amdgcn_wmma_f32_16x16x4_f32(false, a, false, b1, (short)0, acc1, false, false);
        acc2 = __builtin_amdgcn_wmma_f32_16x16x4_f32(false, a, false, b2, (short)0, acc2, false, false);
        acc3 = __builtin_amdgcn_wmma_f32_16x16x4_f32(false, a, false, b3, (short)0, acc3, false, false);
    }

    const float bv0 = bias[j0 +  0 + lm];
    const float bv1 = bias[j0 + 16 + lm];
    const float bv2 = bias[j0 + 32 + lm];
    const float bv3 = bias[j0 + 48 + lm];

    const int bidx = r0 >> 11;        // batch
    const int rr   = r0 & 2047;       // row within batch (t*8+n)
    const int h    = j64;
    #pragma unroll
    for (int r = 0; r < 8; ++r) {
        const int m = r + 8 * lh;     // D row
        float* op = out + ((size_t)(bidx * HEADS + h) * TN + rr + m) * DHEAD + lm;
        op[ 0] = (acc0[r] + bv0) * scale;
        op[16] = (acc1[r] + bv1) * scale;
        op[32] = (acc2[r] + bv2) * scale;
        op[48] = (acc3[r] + bv3) * scale;
    }
}

// ---------------------------------------------------------------------------
// Banded windowed attention. One wave per (b, h, 2-timestep tile):
// 16 query rows x 272-key band, QK^T and P*V both via WMMA f32 16x16x4.
// ---------------------------------------------------------------------------
__global__ __launch_bounds__(32) void attn_kernel(
    const float* __restrict__ ws, const unsigned char* __restrict__ kpm,
    float* __restrict__ out, float* __restrict__ attn)
{
    __shared__ float pl[16 * BAND];   // logits -> probabilities staging

    const int lane = threadIdx.x;
    const int lm = lane & 15;
    const int lh = lane >> 4;

    const int wg = blockIdx.x;            // 0..2047
    const int t0 = (wg & 127) << 1;       // first timestep of this tile
    const int bh = wg >> 7;               // b*8 + h
    const int b  = bh >> 3;
    const int h  = bh & 7;

    const float* qp = ws + (size_t)bh * (TN * DHEAD);
    const float* kp = qp + (size_t)QKV_ELEMS;
    const float* vp = qp + (size_t)2 * QKV_ELEMS;
    const unsigned char* kpmb = kpm + (size_t)b * TN;

    const int r0  = t0 * 8;               // first query row
    const int ks0 = r0 - 128;             // band start key row: 8*(t0-16)

    // Resident Q tile as 16 A-fragments (16x4 slices of the 16x64 Q block)
    v2f aq[16];
    {
        const float* q = qp + (size_t)(r0 + lm) * DHEAD + 2 * lh;
        #pragma unroll
        for (int kk = 0; kk < 16; ++kk) aq[kk] = *(const v2f*)(q + 4 * kk);
    }

    float rm[8];
    #pragma unroll
    for (int r = 0; r < 8; ++r) rm[r] = -3.0e38f;

    // ---- Pass 1: QK^T over the band, mask, stage logits to LDS ----
    for (int j = 0; j < 17; ++j) {
        const int kr0 = ks0 + 16 * j;
        const int krl = kr0 + lm;                         // this lane's key row
        const int krc = krl < 0 ? 0 : (krl > TN - 1 ? TN - 1 : krl);
        const float* krow = kp + (size_t)krc * DHEAD + 2 * lh;

        // Warm this tile's V rows for pass 2 (overlaps with the WMMA stream;
        // emits global_prefetch — gfx1250 VMEM prefetch path).
        {
            const float* vrow = vp + (size_t)krc * DHEAD;
            __builtin_prefetch(vrow, 0, 3);
            __builtin_prefetch(vrow + 32, 0, 3);          // second 128B line
        }

        v8f acc = {};
        #pragma unroll
        for (int kk = 0; kk < 16; ++kk) {
            v2f bf = *(const v2f*)(krow + 4 * kk);
            acc = __builtin_amdgcn_wmma_f32_16x16x4_f32(false, aq[kk], false, bf, (short)0, acc, false, false);
        }

        const bool inb = (krl >= 0) && (krl < TN);
        const bool pad = inb ? (kpmb[krl] != 0) : true;
        #pragma unroll
        for (int r = 0; r < 8; ++r) {
            const int m   = r + 8 * lh;
            const int tq8 = r0 + (m & 8);                 // 8*t_query
            const bool win = (krl >= tq8 - 128) && (krl <= tq8 + 135);
            const float x = (inb && win && !pad) ? acc[r] : -3.0e38f;
            rm[r] = fmaxf(rm[r], x);
            pl[m * BAND + 16 * j + lm] = x;
        }
    }

    // ---- Row max across each 16-lane half (matches D-fragment layout) ----
    #pragma unroll
    for (int r = 0; r < 8; ++r) {
        float v = rm[r];
        v = fmaxf(v, __shfl_xor(v, 1));
        v = fmaxf(v, __shfl_xor(v, 2));
        v = fmaxf(v, __shfl_xor(v, 4));
        v = fmaxf(v, __shfl_xor(v, 8));
        rm[r] = v;
    }

    // ---- exp + row sum, overwrite LDS with unnormalized probabilities ----
    float rs[8];
    #pragma unroll
    for (int r = 0; r < 8; ++r) rs[r] = 0.0f;
    for (int j = 0; j < 17; ++j) {
        #pragma unroll
        for (int r = 0; r < 8; ++r) {
            const int m = r + 8 * lh;
            const float x = pl[m * BAND + 16 * j + lm];
            const float pe = __expf(x - rm[r]);
            rs[r] += pe;
            pl[m * BAND + 16 * j + lm] = pe;
        }
    }
    #pragma unroll
    for (int r = 0; r < 8; ++r) {
        float v = rs[r];
        v += __shfl_xor(v, 1);
        v += __shfl_xor(v, 2);
        v += __shfl_xor(v, 4);
        v += __shfl_xor(v, 8);
        rs[r] = 1.0f / v;
    }

    __syncthreads();
    asm volatile("s_wait_dscnt 0" ::: "memory");   // cross-lane LDS ordering

    // ---- Pass 2: out = P * V over the band ----
    v8f o0 = {}, o1 = {}, o2 = {}, o3 = {};
    for (int j = 0; j < 17; ++j) {
        const int kr0 = ks0 + 16 * j;
        v2f ap[4];
        #pragma unroll
        for (int s = 0; s < 4; ++s)                       // P tile, A layout
            ap[s] = *(const v2f*)&pl[lm * BAND + 16 * j + 4 * s + 2 * lh];
        #pragma unroll
        for (int s = 0; s < 4; ++s) {
            const int vr0 = kr0 + 4 * s + 2 * lh;
            const int va  = vr0     < 0 ? 0 : (vr0     > TN - 1 ? TN - 1 : vr0);
            const int vb  = vr0 + 1 < 0 ? 0 : (vr0 + 1 > TN - 1 ? TN - 1 : vr0 + 1);
            const float* vra = vp + (size_t)va * DHEAD + lm;
            const float* vrb = vp + (size_t)vb * DHEAD + lm;
            v2f b0 = { vra[ 0], vrb[ 0] };
            v2f b1 = { vra[16], vrb[16] };
            v2f b2 = { vra[32], vrb[32] };
            v2f b3 = { vra[48], vrb[48] };
            o0 = __builtin_amdgcn_wmma_f32_16x16x4_f32(false, ap[s], false, b0, (short)0, o0, false, false);
            o1 = __builtin_amdgcn_wmma_f32_16x16x4_f32(false, ap[s], false, b1, (short)0, o1, false, false);
            o2 = __builtin_amdgcn_wmma_f32_16x16x4_f32(false, ap[s], false, b2, (short)0, o2, false, false);
            o3 = __builtin_amdgcn_wmma_f32_16x16x4_f32(false, ap[s], false, b3, (short)0, o3, false, false);
        }
    }

    // ---- Store out[b][t][n][h*64+dc] ----
    #pragma unroll
    for (int r = 0; r < 8; ++r) {
        const int m  = r + 8 * lh;
        const int rq = r0 + m;                            // t*8+n within batch
        float* op = out + ((size_t)b * TN + rq) * EDIM + h * DHEAD + lm;
        const float si = rs[r];
        op[ 0] = o0[r] * si;
        op[16] = o1[r] * si;
        op[32] = o2[r] * si;
        op[48] = o3[r] * si;
    }

    // ---- Store attn[b][t][n][widx][h], widx = band_col - 8*(t-t0) ----
    for (int j = 0; j < 17; ++j) {
        #pragma unroll
        for (int r = 0; r < 8; ++r) {
            const int m = r + 8 * lh;
            const int c = 16 * j + lm;
            const int widx = c - (m & 8);
            if (widx >= 0 && widx < 264) {
                const int rq = r0 + m;
                const float pv = pl[m * BAND + c] * rs[r];
                attn[(((size_t)b * TN + rq) * 264 + widx) * HEADS + h] = pv;
            }
        }
    }
}

// ---------------------------------------------------------------------------
extern "C" void kernel_launch(void* const* d_in, const int* in_sizes, int n_in,
                              void* d_out, int out_size, void* d_ws, size_t ws_size,
                              hipStream_t stream) {
    const float* Xq = (const float*)d_in[0];
    const float* Xk = (const float*)d_in[1];
    const float* Xv = (const float*)d_in[2];
    const unsigned char* kpm = (const unsigned char*)d_in[3];
    const float* Wq = (const float*)d_in[4];
    const float* bq = (const float*)d_in[5];
    const float* Wk = (const float*)d_in[6];
    const float* bk = (const float*)d_in[7];
    const float* Wv = (const float*)d_in[8];
    const float* bv = (const float*)d_in[9];

    float* ws   = (float*)d_ws;
    float* outp = (float*)d_out;
    float* attn = outp + (size_t)BDIM * TDIM * NDIM * EDIM;   // + 2,097,152

    // 3 projections * 256 M-tiles * 8 N64-groups = 6144 waves, 8 waves/block
    proj_kernel<<<768, 256, 0, stream>>>(Xq, Xk, Xv, Wq, bq, Wk, bk, Wv, bv, ws);
    // 2 batches * 8 heads * 128 two-timestep tiles = 2048 single-wave blocks
    attn_kernel<<<2048, 32, 0, stream>>>(ws, kpm, outp, attn);
}